// ContourChamferLoss_2714419331829
// MI455X (gfx1250) — compile-verified
//
#include <hip/hip_runtime.h>
#include <hip/hip_bf16.h>
#include <math.h>

// Vector types matching the CDNA5 WMMA f32 16x16x4 operand shapes (wave32):
//   A (16x4 f32)  = 64 elems / 32 lanes = 2 VGPRs -> v2f
//   B (4x16 f32)  = 64 elems / 32 lanes = 2 VGPRs -> v2f
//   C/D (16x16)   = 256 elems / 32 lanes = 8 VGPRs -> v8f
typedef float v2f __attribute__((ext_vector_type(2)));
typedef float v8f __attribute__((ext_vector_type(8)));

#define SUBSAMPLE 8
#define BIGD 1e30f

// reference: idx = float32 linspace(0, n-1, cnt) truncated to int
__device__ __forceinline__ int sub_idx(int i, int n, int cnt) {
  float step = (float)(n - 1) / (float)(cnt - 1);
  int idx = (int)((float)i * step);
  return idx < (n - 1) ? idx : (n - 1);
}

// Build augmented 4-vectors so that a single V_WMMA_F32_16X16X4_F32 produces
// squared distances directly:
//   mode 0 (render pts, A rows):  (-2x, -2y, x^2+y^2, 1);  pad -> (0,0,BIGD,0)
//   mode 1 (ref pts,   B rows):   ( x,  y,  1, x^2+y^2);   pad -> (0,0,0,BIGD)
__global__ void chamfer_build_aug(const float* __restrict__ src, int npts,
                                  int cnt, int cntPad,
                                  float* __restrict__ dst, int mode) {
  int i = blockIdx.x * blockDim.x + threadIdx.x;
  if (i >= cntPad) return;
  float4 o;
  if (i < cnt) {
    int s = sub_idx(i, npts, cnt);
    float x = src[2 * s + 0];
    float y = src[2 * s + 1];
    float n2 = fmaf(x, x, y * y);
    if (mode == 0) o = make_float4(-2.0f * x, -2.0f * y, n2, 1.0f);
    else           o = make_float4(x, y, 1.0f, n2);
  } else {
    if (mode == 0) o = make_float4(0.0f, 0.0f, BIGD, 0.0f);
    else           o = make_float4(0.0f, 0.0f, 0.0f, BIGD);
  }
  reinterpret_cast<float4*>(dst)[i] = o;
}

// One wave owns one 16-row tile of A; loops all B tiles, one WMMA per tile,
// min-accumulates the 8 accumulator VGPRs, then reduces across the 16-lane
// halves with shfl_xor. Lane layout (ISA 7.12.2): A/B operand k = vgpr + 2*half.
__global__ void __launch_bounds__(256)
chamfer_rowmin(const float* __restrict__ Aaug, const float* __restrict__ Baug,
               int Ntiles, int Mtiles, int N, float* __restrict__ rowmin) {
  int wave = (int)((blockIdx.x * blockDim.x + threadIdx.x) >> 5);
  int lane = (int)(threadIdx.x & 31);
  if (wave >= Ntiles) return;  // wave-uniform: EXEC all-1s inside
  int half = lane >> 4;
  int lm   = lane & 15;
  int kb   = half * 2;  // k-pair base for this lane half

  v2f a = *reinterpret_cast<const v2f*>(Aaug + ((size_t)(wave * 16 + lm) * 4 + kb));

  v8f acc;
#pragma unroll
  for (int v = 0; v < 8; ++v) acc[v] = BIGD;

  for (int j = 0; j < Mtiles; ++j) {
    v2f b = *reinterpret_cast<const v2f*>(Baug + ((size_t)(j * 16 + lm) * 4 + kb));
    v8f c = {};
    v8f d = __builtin_amdgcn_wmma_f32_16x16x4_f32(
        /*neg_a=*/false, a, /*neg_b=*/false, b,
        /*c_mod=*/(short)0, c, /*reuse_a=*/false, /*reuse_b=*/false);
#pragma unroll
    for (int v = 0; v < 8; ++v) acc[v] = fminf(acc[v], d[v]);
  }

  // Reduce across columns (lanes) within each 16-lane half; xor masks <16 stay
  // inside the half. VGPR v holds row M=v (half 0) / M=v+8 (half 1).
#pragma unroll
  for (int v = 0; v < 8; ++v) {
    float r = acc[v];
#pragma unroll
    for (int m = 1; m < 16; m <<= 1) r = fminf(r, __shfl_xor(r, m, 32));
    int row = wave * 16 + half * 8 + v;
    if (lm == 0 && row < N) rowmin[row] = sqrtf(r);
  }
}

// One wave owns one 16-col tile of B; loops all A tiles; final reduce over the
// 16 M-values = tree-min of 8 VGPRs + one half-swap shuffle.
__global__ void __launch_bounds__(256)
chamfer_colmin(const float* __restrict__ Aaug, const float* __restrict__ Baug,
               int Ntiles, int Mtiles, int M, float* __restrict__ colmin) {
  int wave = (int)((blockIdx.x * blockDim.x + threadIdx.x) >> 5);
  int lane = (int)(threadIdx.x & 31);
  if (wave >= Mtiles) return;  // wave-uniform
  int half = lane >> 4;
  int lm   = lane & 15;
  int kb   = half * 2;

  v2f b = *reinterpret_cast<const v2f*>(Baug + ((size_t)(wave * 16 + lm) * 4 + kb));

  v8f acc;
#pragma unroll
  for (int v = 0; v < 8; ++v) acc[v] = BIGD;

  for (int i = 0; i < Ntiles; ++i) {
    v2f a = *reinterpret_cast<const v2f*>(Aaug + ((size_t)(i * 16 + lm) * 4 + kb));
    v8f c = {};
    v8f d = __builtin_amdgcn_wmma_f32_16x16x4_f32(
        false, a, false, b, (short)0, c, false, false);
#pragma unroll
    for (int v = 0; v < 8; ++v) acc[v] = fminf(acc[v], d[v]);
  }

  // min over M = 0..7 (this half's VGPRs), then combine halves (M vs M+8)
  float r = acc[0];
#pragma unroll
  for (int v = 1; v < 8; ++v) r = fminf(r, acc[v]);
  r = fminf(r, __shfl_xor(r, 16, 32));

  int col = wave * 16 + lm;
  if (half == 0 && col < M) colmin[col] = sqrtf(r);
}

__global__ void __launch_bounds__(256)
chamfer_finalize(const float* __restrict__ rowmin, const float* __restrict__ colmin,
                 int N, int M, float* __restrict__ out) {
  __shared__ float s1[256];
  __shared__ float s2[256];
  float a = 0.0f, b = 0.0f;
  for (int i = threadIdx.x; i < N; i += 256) a += rowmin[i];
  for (int i = threadIdx.x; i < M; i += 256) b += colmin[i];
  s1[threadIdx.x] = a;
  s2[threadIdx.x] = b;
  __syncthreads();
  for (int s = 128; s > 0; s >>= 1) {
    if ((int)threadIdx.x < s) {
      s1[threadIdx.x] += s1[threadIdx.x + s];
      s2[threadIdx.x] += s2[threadIdx.x + s];
    }
    __syncthreads();
  }
  if (threadIdx.x == 0)
    out[0] = 0.5f * (s1[0] / (float)N + s2[0] / (float)M);
}

extern "C" void kernel_launch(void* const* d_in, const int* in_sizes, int n_in,
                              void* d_out, int out_size, void* d_ws, size_t ws_size,
                              hipStream_t stream) {
  const float* pc  = (const float*)d_in[0];  // (1000,64,2) flat
  const float* ref = (const float*)d_in[1];  // (80000,2) flat

  int npts0 = in_sizes[0] / 2;              // 64000
  int npts1 = in_sizes[1] / 2;              // 80000
  int N = npts0 / SUBSAMPLE;                // 8000
  int M = npts1 / SUBSAMPLE;                // 10000
  int Ntiles = (N + 15) / 16;               // 500
  int Mtiles = (M + 15) / 16;               // 625
  int Npad = Ntiles * 16;
  int Mpad = Mtiles * 16;

  // workspace layout (floats): Aaug[Npad*4] | Baug[Mpad*4] | rowmin[Npad] | colmin[Mpad]
  float* Aaug   = (float*)d_ws;
  float* Baug   = Aaug + (size_t)Npad * 4;
  float* rowmin = Baug + (size_t)Mpad * 4;
  float* colmin = rowmin + Npad;

  chamfer_build_aug<<<(Npad + 255) / 256, 256, 0, stream>>>(pc, npts0, N, Npad, Aaug, 0);
  chamfer_build_aug<<<(Mpad + 255) / 256, 256, 0, stream>>>(ref, npts1, M, Mpad, Baug, 1);

  int rowThreads = Ntiles * 32;
  chamfer_rowmin<<<(rowThreads + 255) / 256, 256, 0, stream>>>(Aaug, Baug, Ntiles, Mtiles, N, rowmin);

  int colThreads = Mtiles * 32;
  chamfer_colmin<<<(colThreads + 255) / 256, 256, 0, stream>>>(Aaug, Baug, Ntiles, Mtiles, M, colmin);

  chamfer_finalize<<<1, 256, 0, stream>>>(rowmin, colmin, N, M, (float*)d_out);
}